// MPNet_69346541962036
// MI455X (gfx1250) — compile-verified
//
#include <hip/hip_runtime.h>
#include <hip/hip_bf16.h>

// ---------------- problem constants (match reference) ----------------
constexpr int Bx    = 8;
constexpr int WIN   = 12;
constexpr int NN    = 2000;
constexpr int F_IN  = 8;
constexpr int H     = 64;
constexpr int Ntot  = Bx * WIN * NN;        // 192000 graph nodes
constexpr int Etot  = 2000000;              // edges
constexpr int BNN   = Bx * NN;              // 16000 sequences
constexpr int FC1IN = 2 * H + WIN * F_IN + 1; // 225
constexpr int KP    = 232;                  // FC1 K padded to a multiple of 8
constexpr float EPS = 1e-5f;

typedef float v2f __attribute__((ext_vector_type(2)));
typedef float v8f __attribute__((ext_vector_type(8)));

// ---------------- small helpers ----------------
__global__ void k_fill(float* p, float v, long n) {
  long i = (long)blockIdx.x * blockDim.x + threadIdx.x;
  if (i < n) p[i] = v;
}

__global__ void k_transpose(const float* __restrict__ in, float* __restrict__ out,
                            int rows, int cols) {
  int i = blockIdx.x * blockDim.x + threadIdx.x;
  if (i < rows * cols) {
    int r = i / cols, c = i % cols;
    out[(size_t)c * rows + r] = in[i];
  }
}

// fc1_w [225,64] -> zero-padded [232,64]
__global__ void k_pad_fc1w(const float* __restrict__ in, float* __restrict__ out) {
  int i = blockIdx.x * blockDim.x + threadIdx.x;
  if (i < KP * H) {
    int r = i >> 6, c = i & 63;
    out[i] = (r < FC1IN) ? in[r * H + c] : 0.f;
  }
}

// ---------------- degree / symmetric normalization ----------------
__global__ void k_deg(const int* __restrict__ col, const float* __restrict__ w,
                      float* __restrict__ deg) {
  int e = blockIdx.x * blockDim.x + threadIdx.x;
  if (e < Etot) atomicAdd(&deg[col[e]], w[e]);
}

__global__ void k_dis(const float* __restrict__ deg, float* __restrict__ dis) {
  int n = blockIdx.x * blockDim.x + threadIdx.x;
  if (n < Ntot) { float d = deg[n]; dis[n] = d > 0.f ? rsqrtf(d) : 0.f; }
}

// ---------------- fp32 WMMA GEMM: C[M,Nc] (+)= A[M,K] * B[K,Nc] ----------------
// Requirements (guaranteed by callers): M % 16 == 0, Nc % 32 == 0, K % 8 == 0.
// One wave owns a 16x32 C tile (two accumulators sharing the A fragment);
// block = 8 waves stacked along M (128 rows). No guarded loads -> straight-line
// global_load_b64 / b32 batches + 4 back-to-back v_wmma per iteration.
// Fragment layouts per CDNA5 ISA 7.12.2 (32-bit A 16x4 / 32-bit C 16x16).
__global__ __launch_bounds__(256)
void k_wmma_gemm(const float* __restrict__ A, const float* __restrict__ Bm,
                 float* __restrict__ C, int M, int Nc, int K, int lda, int accum)
{
  const int lane  = threadIdx.x & 31;
  const int wave  = threadIdx.x >> 5;
  const int tileM = (blockIdx.x * 8 + wave) * 16;
  const int tileN = blockIdx.y * 32;
  if (tileM >= M) return;

  const int half = lane >> 4;          // lane half selects K pair within fragment
  const int l16  = lane & 15;
  const int am   = tileM + l16;        // A row for this lane
  const int bn0  = tileN + l16;        // first C column tile
  const int bn1  = bn0 + 16;           // second C column tile
  const int koff = half * 2;
  const int cm   = tileM + half * 8;   // C rows cm..cm+7 for this lane half

  v8f c0 = {}, c1 = {};
  if (accum) {
#pragma unroll
    for (int i = 0; i < 8; ++i) {
      c0[i] = C[(size_t)(cm + i) * Nc + bn0];
      c1[i] = C[(size_t)(cm + i) * Nc + bn1];
    }
  }

  const float* Ap = A + (size_t)am * lda + koff;
  for (int k = 0; k < K; k += 8) {
    v2f a0, a1, b00, b01, b10, b11;
    a0.x = Ap[k];     a0.y = Ap[k + 1];       // contiguous -> b64
    a1.x = Ap[k + 4]; a1.y = Ap[k + 5];
    const float* Bp0 = Bm + (size_t)(k + koff) * Nc;
    const float* Bp1 = Bm + (size_t)(k + 4 + koff) * Nc;
    b00.x = Bp0[bn0]; b00.y = Bp0[Nc + bn0];
    b01.x = Bp0[bn1]; b01.y = Bp0[Nc + bn1];
    b10.x = Bp1[bn0]; b10.y = Bp1[Nc + bn0];
    b11.x = Bp1[bn1]; b11.y = Bp1[Nc + bn1];
    c0 = __builtin_amdgcn_wmma_f32_16x16x4_f32(false, a0, false, b00, (short)0, c0, false, false);
    c1 = __builtin_amdgcn_wmma_f32_16x16x4_f32(false, a0, false, b01, (short)0, c1, false, false);
    c0 = __builtin_amdgcn_wmma_f32_16x16x4_f32(false, a1, false, b10, (short)0, c0, false, false);
    c1 = __builtin_amdgcn_wmma_f32_16x16x4_f32(false, a1, false, b11, (short)0, c1, false, false);
  }

#pragma unroll
  for (int i = 0; i < 8; ++i) {
    C[(size_t)(cm + i) * Nc + bn0] = c0[i];
    C[(size_t)(cm + i) * Nc + bn1] = c1[i];
  }
}

// ---------------- GCN aggregation ----------------
// self loop (weight 1, norm = dis[n]^2) + bias
__global__ void k_agg_init(const float* __restrict__ hin, const float* __restrict__ dis,
                           const float* __restrict__ bias, float* __restrict__ agg) {
  long i = (long)blockIdx.x * blockDim.x + threadIdx.x;
  if (i < (long)Ntot * H) {
    int n = (int)(i >> 6), f = (int)(i & 63);
    float s = dis[n];
    agg[i] = hin[i] * s * s + bias[f];
  }
}

// one wave per edge; lanes cover the 64 features (coalesced gather + atomic scatter)
__global__ __launch_bounds__(256)
void k_agg_edges(const int* __restrict__ row, const int* __restrict__ col,
                 const float* __restrict__ w, const float* __restrict__ dis,
                 const float* __restrict__ hin, float* __restrict__ agg) {
  long t = (long)blockIdx.x * blockDim.x + threadIdx.x;
  int e = (int)(t >> 5), lane = (int)(t & 31);
  if (e >= Etot) return;
  int r = row[e], cn = col[e];
  float nrm = dis[r] * w[e] * dis[cn];
  const float* hs = hin + (size_t)r * H;
  float* ad = agg + (size_t)cn * H;
  atomicAdd(&ad[lane],      hs[lane]      * nrm);
  atomicAdd(&ad[lane + 32], hs[lane + 32] * nrm);
}

// ---------------- fused ReLU + BN statistics (sum / sumsq per feature) ----------------
__global__ __launch_bounds__(256)
void k_relu_stats(float* __restrict__ h, float* __restrict__ stats) {
  __shared__ float ssum[256], ssq[256];
  const long n = (long)Ntot * H;
  const long stride = (long)gridDim.x * blockDim.x;   // multiple of 64
  float s = 0.f, sq = 0.f;
  for (long i = (long)blockIdx.x * blockDim.x + threadIdx.x; i < n; i += stride) {
    float v = h[i]; v = v > 0.f ? v : 0.f; h[i] = v;
    s += v; sq += v * v;
  }
  ssum[threadIdx.x] = s; ssq[threadIdx.x] = sq;
  __syncthreads();
  if (threadIdx.x < 64) {
    int f = threadIdx.x;
    atomicAdd(&stats[f],      ssum[f] + ssum[f + 64] + ssum[f + 128] + ssum[f + 192]);
    atomicAdd(&stats[64 + f], ssq[f]  + ssq[f + 64]  + ssq[f + 128]  + ssq[f + 192]);
  }
}

__global__ void k_bn(float* __restrict__ h, const float* __restrict__ stats,
                     const float* __restrict__ g, const float* __restrict__ be) {
  long i = (long)blockIdx.x * blockDim.x + threadIdx.x;
  if (i < (long)Ntot * H) {
    int f = (int)(i & 63);
    float m = stats[f] * (1.f / Ntot);
    float v = stats[64 + f] * (1.f / Ntot) - m * m;   // biased variance
    h[i] = (h[i] - m) * rsqrtf(v + EPS) * g[f] + be[f];
  }
}

// ---------------- LSTM1 input gather: xt[r, 0:128] = concat(h1,h2)[node(r,t)] ----------------
__global__ void k_gather_xt(const float* __restrict__ h1, const float* __restrict__ h2,
                            float* __restrict__ xt, int t) {
  long i = (long)blockIdx.x * blockDim.x + threadIdx.x;
  if (i >= (long)BNN * 2 * H) return;
  int r = (int)(i >> 7), f = (int)(i & 127);
  int b = r / NN, nn = r % NN;
  long node = ((long)b * WIN + t) * NN + nn;
  xt[i] = (f < H) ? h1[node * H + f] : h2[node * H + (f - H)];
}

// ---------------- fused LSTM cell (gates pre-computed, torch i,f,g,o order) ----------------
__global__ void k_lstm_cell(const float* __restrict__ gates, const float* __restrict__ bih,
                            const float* __restrict__ bhh, float* __restrict__ hbuf,
                            float* __restrict__ cbuf, float* __restrict__ yout) {
  long i = (long)blockIdx.x * blockDim.x + threadIdx.x;
  if (i >= (long)BNN * H) return;
  int r = (int)(i >> 6), f = (int)(i & 63);
  const float* gr = gates + (size_t)r * 4 * H;
  float ig = gr[f]         + bih[f]         + bhh[f];
  float fg = gr[H + f]     + bih[H + f]     + bhh[H + f];
  float gg = gr[2 * H + f] + bih[2 * H + f] + bhh[2 * H + f];
  float og = gr[3 * H + f] + bih[3 * H + f] + bhh[3 * H + f];
  float si = 1.f / (1.f + expf(-ig));
  float sf = 1.f / (1.f + expf(-fg));
  float so = 1.f / (1.f + expf(-og));
  float c  = sf * cbuf[i] + si * tanhf(gg);
  float hh = so * tanhf(c);
  cbuf[i] = c; hbuf[i] = hh;
  if (yout) yout[i] = hh;
}

// ---------------- FC head ----------------
// z row stride KP=232; cols 225..231 zero-padded (so FC1 GEMM needs no K guard)
__global__ void k_assemble_z(const float* __restrict__ hn1, const float* __restrict__ hn2,
                             const float* __restrict__ inner, const float* __restrict__ x,
                             float* __restrict__ z) {
  long i = (long)blockIdx.x * blockDim.x + threadIdx.x;
  if (i >= (long)BNN * KP) return;
  int r = (int)(i / KP), j = (int)(i % KP);
  float v = 0.f;
  if (j < H)               v = hn1[(size_t)r * H + j];
  else if (j < 2 * H)      v = hn2[(size_t)r * H + (j - H)];
  else if (j == 2 * H)     v = inner[r];
  else if (j < FC1IN) {
    int j2 = j - (2 * H + 1);
    int wi = j2 / F_IN, f = j2 % F_IN;
    int b = r / NN, nn = r % NN;
    long node = ((long)b * WIN + wi) * NN + nn;
    v = x[node * F_IN + f];
  }
  z[i] = v;
}

__global__ void k_bias_relu(float* __restrict__ h, const float* __restrict__ bias,
                            long n, int nc) {
  long i = (long)blockIdx.x * blockDim.x + threadIdx.x;
  if (i < n) {
    int f = (int)(i % nc);
    float v = h[i] + bias[f];
    h[i] = v > 0.f ? v : 0.f;
  }
}

// one wave per row: 64-wide dot with fc3_w, butterfly reduce, relu
__global__ __launch_bounds__(256)
void k_fc3(const float* __restrict__ f1, const float* __restrict__ w3,
           const float* __restrict__ b3, float* __restrict__ out) {
  int gid = blockIdx.x * blockDim.x + threadIdx.x;
  int r = gid >> 5, lane = gid & 31;
  if (r >= BNN) return;
  const float* fr = f1 + (size_t)r * H;
  float s = fr[lane] * w3[lane] + fr[lane + 32] * w3[lane + 32];
  for (int m = 16; m; m >>= 1) s += __shfl_xor(s, m, 32);
  if (lane == 0) { float v = s + b3[0]; out[r] = v > 0.f ? v : 0.f; }
}

// ---------------- host-side orchestration ----------------
static inline int cdiv(long a, int b) { return (int)((a + b - 1) / b); }

extern "C" void kernel_launch(void* const* d_in, const int* in_sizes, int n_in,
                              void* d_out, int out_size, void* d_ws, size_t ws_size,
                              hipStream_t stream) {
  (void)in_sizes; (void)n_in; (void)out_size; (void)ws_size;

  const int*   adj   = (const int*)d_in[0];
  const int*   row   = adj;
  const int*   col   = adj + Etot;
  const float* aw    = (const float*)d_in[1];
  const float* x     = (const float*)d_in[2];
  const float* inner = (const float*)d_in[3];
  const float* W1    = (const float*)d_in[4];
  const float* b1    = (const float*)d_in[5];
  const float* W2    = (const float*)d_in[6];
  const float* b2    = (const float*)d_in[7];
  const float* g1    = (const float*)d_in[8];
  const float* be1   = (const float*)d_in[9];
  const float* g2    = (const float*)d_in[10];
  const float* be2   = (const float*)d_in[11];
  const float* Wih1  = (const float*)d_in[12];
  const float* Whh1  = (const float*)d_in[13];
  const float* bih1  = (const float*)d_in[14];
  const float* bhh1  = (const float*)d_in[15];
  const float* Wih2  = (const float*)d_in[16];
  const float* Whh2  = (const float*)d_in[17];
  const float* bih2  = (const float*)d_in[18];
  const float* bhh2  = (const float*)d_in[19];
  const float* fc1w  = (const float*)d_in[20];
  const float* fc1b  = (const float*)d_in[21];
  const float* fc3w  = (const float*)d_in[22];
  const float* fc3b  = (const float*)d_in[23];

  // workspace layout (floats) -- ~235 MB total
  const size_t NH = (size_t)Ntot * H;          // 12.288M
  float* ws    = (float*)d_ws;
  float* deg   = ws;                           // Ntot
  float* dis   = deg + Ntot;                   // Ntot
  float* tmp   = dis + Ntot;                   // NH  (x*W1 / h1*W2; later xt+gates)
  float* h1    = tmp + NH;                     // NH
  float* h2    = h1 + NH;                      // NH
  float* y1    = h2 + NH;                      // WIN*BNN*H = 12.288M
  float* st    = y1 + (size_t)WIN * BNN * H;   // 128 BN stats
  float* wT1i  = st + 128;                     // Wih1^T [128,256]
  float* wT1h  = wT1i + 2 * H * 4 * H;         // Whh1^T [64,256]
  float* wT2i  = wT1h + H * 4 * H;             // Wih2^T [64,256]
  float* wT2h  = wT2i + H * 4 * H;             // Whh2^T [64,256]
  float* fc1wp = wT2h + H * 4 * H;             // padded fc1_w [232,64]
  float* hs1   = fc1wp + KP * H;               // BNN*H (ends as hn1)
  float* cs1   = hs1 + (size_t)BNN * H;
  float* hs2   = cs1 + (size_t)BNN * H;        // ends as hn2
  float* cs2   = hs2 + (size_t)BNN * H;
  float* z     = cs2 + (size_t)BNN * H;        // BNN*KP
  float* f1    = z + (size_t)BNN * KP;         // BNN*H
  // xt / gates alias the (then-free) tmp region during the LSTM phase
  float* xt    = tmp;                          // BNN*128
  float* gates = tmp + (size_t)BNN * 2 * H;    // BNN*256

  const long nh = (long)Ntot * H;

  // ---- degree + norm ----
  k_fill<<<cdiv(Ntot, 256), 256, 0, stream>>>(deg, 1.0f, Ntot);   // self loops
  k_deg<<<cdiv(Etot, 256), 256, 0, stream>>>(col, aw, deg);
  k_dis<<<cdiv(Ntot, 256), 256, 0, stream>>>(deg, dis);

  // ---- weight prep: transposes to [K,N] row-major, padded fc1_w ----
  k_transpose<<<cdiv(4 * H * 2 * H, 256), 256, 0, stream>>>(Wih1, wT1i, 4 * H, 2 * H);
  k_transpose<<<cdiv(4 * H * H, 256), 256, 0, stream>>>(Whh1, wT1h, 4 * H, H);
  k_transpose<<<cdiv(4 * H * H, 256), 256, 0, stream>>>(Wih2, wT2i, 4 * H, H);
  k_transpose<<<cdiv(4 * H * H, 256), 256, 0, stream>>>(Whh2, wT2h, 4 * H, H);
  k_pad_fc1w<<<cdiv(KP * H, 256), 256, 0, stream>>>(fc1w, fc1wp);

  // ---- GCN layer 1 ----
  k_wmma_gemm<<<dim3(Ntot / 128, H / 32), 256, 0, stream>>>(x, W1, tmp, Ntot, H, F_IN, F_IN, 0);
  k_agg_init<<<cdiv(nh, 256), 256, 0, stream>>>(tmp, dis, b1, h1);
  k_agg_edges<<<cdiv((long)Etot * 32, 256), 256, 0, stream>>>(row, col, aw, dis, tmp, h1);
  k_fill<<<1, 128, 0, stream>>>(st, 0.f, 128);
  k_relu_stats<<<512, 256, 0, stream>>>(h1, st);
  k_bn<<<cdiv(nh, 256), 256, 0, stream>>>(h1, st, g1, be1);

  // ---- GCN layer 2 ----
  k_wmma_gemm<<<dim3(Ntot / 128, H / 32), 256, 0, stream>>>(h1, W2, tmp, Ntot, H, H, H, 0);
  k_agg_init<<<cdiv(nh, 256), 256, 0, stream>>>(tmp, dis, b2, h2);
  k_agg_edges<<<cdiv((long)Etot * 32, 256), 256, 0, stream>>>(row, col, aw, dis, tmp, h2);
  k_fill<<<1, 128, 0, stream>>>(st, 0.f, 128);
  k_relu_stats<<<512, 256, 0, stream>>>(h2, st);
  k_bn<<<cdiv(nh, 256), 256, 0, stream>>>(h2, st, g2, be2);

  // ---- LSTM 1 (input 2H=128, hidden 64, T=12, batch 16000) ----
  k_fill<<<cdiv((long)BNN * H, 256), 256, 0, stream>>>(hs1, 0.f, (long)BNN * H);
  k_fill<<<cdiv((long)BNN * H, 256), 256, 0, stream>>>(cs1, 0.f, (long)BNN * H);
  for (int t = 0; t < WIN; ++t) {
    k_gather_xt<<<cdiv((long)BNN * 2 * H, 256), 256, 0, stream>>>(h1, h2, xt, t);
    k_wmma_gemm<<<dim3(BNN / 128, 4 * H / 32), 256, 0, stream>>>(xt, wT1i, gates,
                                                                 BNN, 4 * H, 2 * H, 2 * H, 0);
    k_wmma_gemm<<<dim3(BNN / 128, 4 * H / 32), 256, 0, stream>>>(hs1, wT1h, gates,
                                                                 BNN, 4 * H, H, H, 1);
    k_lstm_cell<<<cdiv((long)BNN * H, 256), 256, 0, stream>>>(gates, bih1, bhh1, hs1, cs1,
                                                              y1 + (size_t)t * BNN * H);
  }

  // ---- LSTM 2 (input H, hidden 64) ----
  k_fill<<<cdiv((long)BNN * H, 256), 256, 0, stream>>>(hs2, 0.f, (long)BNN * H);
  k_fill<<<cdiv((long)BNN * H, 256), 256, 0, stream>>>(cs2, 0.f, (long)BNN * H);
  for (int t = 0; t < WIN; ++t) {
    const float* xt2 = y1 + (size_t)t * BNN * H;
    k_wmma_gemm<<<dim3(BNN / 128, 4 * H / 32), 256, 0, stream>>>(xt2, wT2i, gates,
                                                                 BNN, 4 * H, H, H, 0);
    k_wmma_gemm<<<dim3(BNN / 128, 4 * H / 32), 256, 0, stream>>>(hs2, wT2h, gates,
                                                                 BNN, 4 * H, H, H, 1);
    k_lstm_cell<<<cdiv((long)BNN * H, 256), 256, 0, stream>>>(gates, bih2, bhh2, hs2, cs2,
                                                              nullptr);
  }

  // ---- FC head ----
  k_assemble_z<<<cdiv((long)BNN * KP, 256), 256, 0, stream>>>(hs1, hs2, inner, x, z);
  k_wmma_gemm<<<dim3(BNN / 128, H / 32), 256, 0, stream>>>(z, fc1wp, f1,
                                                           BNN, H, KP, KP, 0);
  k_bias_relu<<<cdiv((long)BNN * H, 256), 256, 0, stream>>>(f1, fc1b, (long)BNN * H, H);
  k_fc3<<<cdiv((long)BNN * 32, 256), 256, 0, stream>>>(f1, fc3w, fc3b, (float*)d_out);
}